// GraphLayer_55233279427254
// MI455X (gfx1250) — compile-verified
//
#include <hip/hip_runtime.h>
#include <cstdint>

#define D_FEAT 128
#define WAVES_PER_BLOCK 8
#define BLOCK_THREADS 256
#define EDGES_PER_WAVE 4

// ---------------------------------------------------------------------------
// Kernel 1: zero the accumulator (d_out is poisoned by the harness).
// ---------------------------------------------------------------------------
__global__ __launch_bounds__(BLOCK_THREADS) void zero_out_kernel(float4* __restrict__ out,
                                                                 int n4) {
  int i = blockIdx.x * blockDim.x + threadIdx.x;
  if (i < n4) {
    out[i] = make_float4(0.f, 0.f, 0.f, 0.f);
  }
}

// ---------------------------------------------------------------------------
// Per-edge scatter: out[d] += a * (x * w), lane l owns features [4l, 4l+4).
// ---------------------------------------------------------------------------
__device__ __forceinline__ void scatter_edge(float* __restrict__ out, int d, float a,
                                             const float4& x, const float4& wv, int lane) {
  float* o = out + (size_t)d * D_FEAT + lane * 4;
  atomicAdd(o + 0, a * x.x * wv.x);
  atomicAdd(o + 1, a * x.y * wv.y);
  atomicAdd(o + 2, a * x.z * wv.z);
  atomicAdd(o + 3, a * x.w * wv.w);
}

// ---------------------------------------------------------------------------
// Kernel 2: fused scatter  out[dst] += a_e * (X[src] * w)
//   - 4 consecutive edges per wave32: scalar metadata loads clause together,
//     4 independent 128-bit gathers in flight per wave (MLP vs random L2 hits)
//   - edge base forced wave-uniform via readfirstlane -> SGPR loads
//   - w (512 B) staged into LDS once per block via CDNA5 async global->LDS DMA
//     (ASYNCcnt path), read once per wave into registers
//   - native fp32 atomics into the L2-resident output
// ---------------------------------------------------------------------------
__global__ __launch_bounds__(BLOCK_THREADS) void spmm_scatter_kernel(
    const float* __restrict__ X,     // [N, 128]
    const int* __restrict__ esrc,    // [E]
    const int* __restrict__ edst,    // [E]
    const float* __restrict__ vals,  // [E]
    const float* __restrict__ w,     // [128]
    float* __restrict__ out,         // [N, 128]
    int n_edges) {
  __shared__ __align__(16) float w_lds[D_FEAT];

  const int lane = threadIdx.x & 31;
  const int wave = threadIdx.x >> 5;

  // --- Stage w into LDS: wave 0, each of 32 lanes async-copies 16 B. ---
  if (wave == 0) {
    uint32_t lds_addr = (uint32_t)(uintptr_t)(&w_lds[0]) + (uint32_t)(lane * 16);
    uint64_t gaddr = (uint64_t)(uintptr_t)w + (uint64_t)(lane * 16);
    asm volatile(
        "global_load_async_to_lds_b128 %0, %1, off\n\t"
        "s_wait_asynccnt 0"
        :
        : "v"(lds_addr), "v"(gaddr)
        : "memory");
  }
  __syncthreads();

  const float4 wv = *(const float4*)(&w_lds[lane * 4]);

  int base = (blockIdx.x * WAVES_PER_BLOCK + wave) * EDGES_PER_WAVE;
  base = __builtin_amdgcn_readfirstlane(base);
  if (base >= n_edges) return;

  const size_t loff = (size_t)(lane * 4);

  if (base + EDGES_PER_WAVE <= n_edges) {
    // Fast path: all metadata in one scalar clause, all gathers in flight.
    const int s0 = esrc[base + 0], s1 = esrc[base + 1], s2 = esrc[base + 2],
              s3 = esrc[base + 3];
    const int d0 = edst[base + 0], d1 = edst[base + 1], d2 = edst[base + 2],
              d3 = edst[base + 3];
    const float a0 = vals[base + 0], a1 = vals[base + 1], a2 = vals[base + 2],
                a3 = vals[base + 3];

    const float4 x0 = *(const float4*)(X + (size_t)s0 * D_FEAT + loff);
    const float4 x1 = *(const float4*)(X + (size_t)s1 * D_FEAT + loff);
    const float4 x2 = *(const float4*)(X + (size_t)s2 * D_FEAT + loff);
    const float4 x3 = *(const float4*)(X + (size_t)s3 * D_FEAT + loff);

    scatter_edge(out, d0, a0, x0, wv, lane);
    scatter_edge(out, d1, a1, x1, wv, lane);
    scatter_edge(out, d2, a2, x2, wv, lane);
    scatter_edge(out, d3, a3, x3, wv, lane);
  } else {
    // Tail path (unused for E % 32 == 0, kept for generality).
    for (int j = 0; j < EDGES_PER_WAVE; ++j) {
      const int e = base + j;
      if (e >= n_edges) break;
      const int s = esrc[e];
      const int d = edst[e];
      const float a = vals[e];
      const float4 x = *(const float4*)(X + (size_t)s * D_FEAT + loff);
      scatter_edge(out, d, a, x, wv, lane);
    }
  }
}

// ---------------------------------------------------------------------------
// Launch
// inputs: 0=feature_matrix [N*128] f32, 1=edge_src [E] i32, 2=edge_dst [E] i32,
//         3=adj_vals [E] f32, 4=w [128] f32
// output: z [N*128] f32
// ---------------------------------------------------------------------------
extern "C" void kernel_launch(void* const* d_in, const int* in_sizes, int n_in,
                              void* d_out, int out_size, void* d_ws, size_t ws_size,
                              hipStream_t stream) {
  const float* X = (const float*)d_in[0];
  const int* esrc = (const int*)d_in[1];
  const int* edst = (const int*)d_in[2];
  const float* vals = (const float*)d_in[3];
  const float* w = (const float*)d_in[4];
  float* out = (float*)d_out;

  const int n_edges = in_sizes[1];  // 640000
  const int n4 = out_size / 4;      // 6.4M floats -> 1.6M float4

  zero_out_kernel<<<(n4 + BLOCK_THREADS - 1) / BLOCK_THREADS, BLOCK_THREADS, 0, stream>>>(
      (float4*)out, n4);

  const int edges_per_block = WAVES_PER_BLOCK * EDGES_PER_WAVE;  // 32
  const int blocks = (n_edges + edges_per_block - 1) / edges_per_block;  // 20000
  spmm_scatter_kernel<<<blocks, BLOCK_THREADS, 0, stream>>>(X, esrc, edst, vals, w, out,
                                                            n_edges);
}